// MeshTransformer_75522704932956
// MI455X (gfx1250) — compile-verified
//
#include <hip/hip_runtime.h>
#include <hip/hip_bf16.h>
#include <math.h>

#define BB 8
#define SS 16
#define PP 32
#define NN 2048
#define VV 2562
#define KS 500
#define COLS (SS*KS)   // 8000

typedef __attribute__((ext_vector_type(2))) float v2f;
typedef __attribute__((ext_vector_type(8))) float v8f;

__device__ __forceinline__ unsigned umin32(unsigned a, unsigned b) { return a < b ? a : b; }
__device__ __forceinline__ unsigned umax32(unsigned a, unsigned b) { return a > b ? a : b; }

// ---------------- init: column minima to +inf ----------------
__global__ void init_colmin(unsigned* __restrict__ colmin, int n) {
    int i = blockIdx.x * blockDim.x + threadIdx.x;
    if (i < n) colmin[i] = 0x7F800000u;   // +inf bits
}

// ---------------- predicted points + centroids ----------------
__global__ __launch_bounds__(256) void points_kernel(
    const float* __restrict__ scales, const float* __restrict__ transforms,
    const float* __restrict__ pw,     const float* __restrict__ off,
    const float* __restrict__ verts,  float* __restrict__ pred,
    float* __restrict__ cent)
{
    __shared__ float Ml[PP][9];
    __shared__ float wt[PP][3];
    __shared__ float tb[3];
    __shared__ float red[256][3];

    const int bs  = blockIdx.x;          // b*S + s
    const int tid = threadIdx.x;

    if (tid < PP) {
        const int p = tid;
        const float* tr = transforms + ((size_t)bs * PP + p) * 6;
        const float tx = tr[0], ty = tr[1], tz = tr[2];
        const float a = tr[3], b = tr[4], c = tr[5];
        const float w   = pw[bs * PP + p];
        const float scl = scales[bs];
        const float ca = cosf(a), sa = sinf(a);
        const float cb = cosf(b), sb = sinf(b);
        const float cc = cosf(c), sc = sinf(c);
        // R = Rx @ Ry @ Rz
        const float r00 =  cb*cc,             r01 = -cb*sc,             r02 =  sb;
        const float r10 =  sa*sb*cc + ca*sc,  r11 = -sa*sb*sc + ca*cc,  r12 = -sa*cb;
        const float r20 = -ca*sb*cc + sa*sc,  r21 =  ca*sb*sc + sa*cc,  r22 =  ca*cb;
        const float k = w * scl;
        Ml[p][0]=k*r00; Ml[p][1]=k*r01; Ml[p][2]=k*r02;
        Ml[p][3]=k*r10; Ml[p][4]=k*r11; Ml[p][5]=k*r12;
        Ml[p][6]=k*r20; Ml[p][7]=k*r21; Ml[p][8]=k*r22;
        wt[p][0]=w*tx;  wt[p][1]=w*ty;  wt[p][2]=w*tz;
    }
    __syncthreads();
    if (tid < 3) {
        float s = 0.f;
        for (int p = 0; p < PP; ++p) s += wt[p][tid];
        tb[tid] = s;
    }
    __syncthreads();

    float csx = 0.f, csy = 0.f, csz = 0.f;
    for (int v = tid; v < VV; v += 256) {
        const float vx = verts[v*3+0], vy = verts[v*3+1], vz = verts[v*3+2];
        float fx = tb[0], fy = tb[1], fz = tb[2];
#pragma unroll 4
        for (int p = 0; p < PP; ++p) {
            const float* o = off + ((size_t)p * VV + v) * 3;
            const float dx = vx + o[0], dy = vy + o[1], dz = vz + o[2];
            const float* M = Ml[p];
            fx += M[0]*dx + M[1]*dy + M[2]*dz;
            fy += M[3]*dx + M[4]*dy + M[5]*dz;
            fz += M[6]*dx + M[7]*dy + M[8]*dz;
        }
        if (v < KS) {
            float* pr = pred + ((size_t)bs * KS + v) * 3;
            pr[0] = fx; pr[1] = fy; pr[2] = fz;
        }
        csx += fx; csy += fy; csz += fz;
    }
    red[tid][0]=csx; red[tid][1]=csy; red[tid][2]=csz;
    __syncthreads();
    for (int o = 128; o > 0; o >>= 1) {
        if (tid < o) {
            red[tid][0]+=red[tid+o][0];
            red[tid][1]+=red[tid+o][1];
            red[tid][2]+=red[tid+o][2];
        }
        __syncthreads();
    }
    if (tid == 0) {
        cent[bs*3+0] = red[0][0] / (float)VV;
        cent[bs*3+1] = red[0][1] / (float)VV;
        cent[bs*3+2] = red[0][2] / (float)VV;
    }
}

// ---------------- fused WMMA distance + top3-per-row + min-per-col ----------------
// grid = (N/128, B); 8 waves/block; each wave owns 16 target rows, sweeps 8000 cols.
// All post-clamp comparisons run in the unsigned-integer domain (f32 >= 0 bit
// patterns are order-isomorphic to u32): v_min_u32/v_max_u32 need no NaN
// canonicalization, and cmin is already in bits form for ds_min_u32.
__global__ __launch_bounds__(256) void dist_kernel(
    const float* __restrict__ target, const float* __restrict__ pred,
    unsigned* __restrict__ colmin,    float* __restrict__ rowsum)
{
    __shared__ unsigned t3[8][16][16][3];   // 24 KB: [wave][row][col-lane][top3]
    __shared__ unsigned colm[COLS];         // 32 KB: per-WG column minima (f32 bits)

    const int b    = blockIdx.y;
    const int rb   = blockIdx.x;
    const int tid  = threadIdx.x;
    const int wid  = tid >> 5;
    const int lane = tid & 31;
    const int half = lane >> 4;
    const int li   = lane & 15;
    const int row0 = rb * 128 + wid * 16;
    const int row  = row0 + li;

    for (int i = tid; i < COLS; i += 256) colm[i] = 0x7F800000u;
    __syncthreads();

    const float* X = target + (size_t)b * NN * 3;
    const float* Y = pred   + (size_t)b * COLS * 3;

    const float x0 = X[row*3+0], x1 = X[row*3+1], x2 = X[row*3+2];
    const float xsq = x0*x0 + x1*x1 + x2*x2;

    // A operand: 16x4 f32, rows = targets, K = [x,y,z,1]
    v2f a;
    a.x = half ? x2   : x0;
    a.y = half ? 1.0f : x1;

    // loop-invariant |x|^2 per C-register slot (M = r + 8*half)
    float xs[8];
#pragma unroll
    for (int r = 0; r < 8; ++r)
        xs[r] = __shfl(xsq, r + 8 * half, 32);

    const unsigned UINF = 0x7F800000u;
    unsigned t0[8], t1[8], t2[8];
#pragma unroll
    for (int r = 0; r < 8; ++r) { t0[r]=UINF; t1[r]=UINF; t2[r]=UINF; }

#pragma unroll 2
    for (int ct = 0; ct < COLS/16; ++ct) {
        const int col = ct*16 + li;
        const float y0 = Y[col*3+0], y1 = Y[col*3+1], y2 = Y[col*3+2];
        const float ysq = y0*y0 + y1*y1 + y2*y2;
        // B operand: 4x16 f32, cols = preds, K = [-2y0,-2y1,-2y2,|y|^2]
        v2f bv;
        bv.x = half ? (-2.0f*y2) : (-2.0f*y0);
        bv.y = half ? ysq        : (-2.0f*y1);

        v8f c = {};
        c = __builtin_amdgcn_wmma_f32_16x16x4_f32(
                false, a, false, bv, (short)0, c, false, false);

        unsigned cmin = UINF;
#pragma unroll
        for (int r = 0; r < 8; ++r) {
            const unsigned d2 = __float_as_uint(fmaxf(c[r] + xs[r], 0.0f));
            cmin = umin32(cmin, d2);
            // branchless sorted-3 insertion: 5 integer min/max
            const unsigned lo  = umin32(t0[r], d2);
            const unsigned mid = umax32(t0[r], d2);
            t0[r] = lo;
            const unsigned m1  = umin32(t1[r], mid);
            const unsigned hi  = umax32(t1[r], mid);
            t1[r] = m1;
            t2[r] = umin32(t2[r], hi);
        }
        // all 32 lanes: no-return DS atomic min (half-pair lanes share addr)
        atomicMin(&colm[col], cmin);
    }

    // stash per-lane top3 candidates, then merge 48 -> 3 per row (integer domain)
#pragma unroll
    for (int r = 0; r < 8; ++r) {
        const int m = r + 8 * half;
        t3[wid][m][li][0] = t0[r];
        t3[wid][m][li][1] = t1[r];
        t3[wid][m][li][2] = t2[r];
    }
    __syncthreads();
    if (half == 0) {
        const int m = li;
        unsigned b0 = UINF, b1 = UINF, b2 = UINF;
        for (int j = 0; j < 16; ++j) {
#pragma unroll
            for (int k = 0; k < 3; ++k) {
                const unsigned v   = t3[wid][m][j][k];
                const unsigned lo  = umin32(b0, v);
                const unsigned mid = umax32(b0, v);
                b0 = lo;
                const unsigned m1  = umin32(b1, mid);
                const unsigned hi  = umax32(b1, mid);
                b1 = m1;
                b2 = umin32(b2, hi);
            }
        }
        rowsum[(size_t)b * NN + row0 + m] =
            __uint_as_float(b0) + __uint_as_float(b1) + __uint_as_float(b2);
    }

    // flush WG-local column minima to global (one atomic per column per WG)
    for (int i = tid; i < COLS; i += 256)
        atomicMin(&colmin[(size_t)b * COLS + i], colm[i]);
}

// ---------------- final scalar: combine three losses ----------------
__global__ __launch_bounds__(256) void finish_kernel(
    const float* __restrict__ rowsum, const float* __restrict__ colminf,
    const float* __restrict__ cent,   float* __restrict__ out)
{
    __shared__ float red[3*256];
    const int tid = threadIdx.x;
    float s1 = 0.f, s2 = 0.f, s3 = 0.f;
    for (int i = tid; i < BB*NN;   i += 256) s1 += rowsum[i];
    for (int i = tid; i < BB*COLS; i += 256) s2 += colminf[i];
    for (int t = tid; t < BB*SS*SS; t += 256) {
        const int b = t / (SS*SS);
        const int i = (t / SS) % SS;
        const int j = t % SS;
        if (i != j) {
            const float* ci = cent + (b*SS + i)*3;
            const float* cj = cent + (b*SS + j)*3;
            const float dx = ci[0]-cj[0], dy = ci[1]-cj[1], dz = ci[2]-cj[2];
            const float d = sqrtf(dx*dx + dy*dy + dz*dz);
            float m = 0.5f - d;
            m = m > 0.f ? m : 0.f;
            s3 += expf(5.0f * m);
        }
    }
    red[tid]=s1; red[256+tid]=s2; red[512+tid]=s3;
    __syncthreads();
    for (int o = 128; o > 0; o >>= 1) {
        if (tid < o) {
            red[tid]     += red[tid+o];
            red[256+tid] += red[256+tid+o];
            red[512+tid] += red[512+tid+o];
        }
        __syncthreads();
    }
    if (tid == 0) {
        const float g  = red[0]   / (float)(BB*NN*3);
        const float sl = red[256] / (float)(BB*COLS);
        const float rp = red[512] / (float)(BB*SS*(SS-1));
        out[0] = 0.7f*g + 0.3f*sl + 0.2f*rp;
    }
}

extern "C" void kernel_launch(void* const* d_in, const int* in_sizes, int n_in,
                              void* d_out, int out_size, void* d_ws, size_t ws_size,
                              hipStream_t stream) {
    const float* scales     = (const float*)d_in[0];
    const float* transforms = (const float*)d_in[1];
    const float* pw         = (const float*)d_in[2];
    const float* off        = (const float*)d_in[3];
    const float* target     = (const float*)d_in[4];
    const float* verts      = (const float*)d_in[5];

    float* ws      = (float*)d_ws;
    float* pred    = ws;                        // B*S*KS*3 = 192000 floats
    float* cent    = pred + (size_t)BB*SS*KS*3; // 384
    float* rowsum  = cent + (size_t)BB*SS*3;    // B*N = 16384
    unsigned* colmin = (unsigned*)(rowsum + (size_t)BB*NN); // B*COLS = 64000

    init_colmin<<<(BB*COLS + 255)/256, 256, 0, stream>>>(colmin, BB*COLS);
    points_kernel<<<BB*SS, 256, 0, stream>>>(scales, transforms, pw, off, verts, pred, cent);
    dist_kernel<<<dim3(NN/128, BB), 256, 0, stream>>>(target, pred, colmin, rowsum);
    finish_kernel<<<1, 256, 0, stream>>>(rowsum, (const float*)colmin, cent, (float*)d_out);
}